// GAT_25460566131067
// MI455X (gfx1250) — compile-verified
//
#include <hip/hip_runtime.h>
#include <math.h>
#include <stdint.h>

#define NUM_GRAPHS 512

typedef __attribute__((ext_vector_type(2))) float v2f;
typedef __attribute__((ext_vector_type(8))) float v8f;

__device__ __forceinline__ float warp_sum32(float v) {
#pragma unroll
  for (int off = 16; off > 0; off >>= 1) v += __shfl_xor(v, off, 32);
  return v;
}

// float atomic-max via sign-split monotonic bit patterns:
//  - nonneg floats: int compare order == float order        -> atomicMax(int)
//  - neg floats:    uint compare order reverses float order -> atomicMin(uint)
__device__ __forceinline__ void atomic_max_f32(float* addr, float v) {
  if (v >= 0.0f) atomicMax((int*)addr, __float_as_int(v));
  else           atomicMin((unsigned int*)addr, __float_as_uint(v));
}

// ---------------------------------------------------------------------------
// H[N,64] = A[N,64] @ W[64,64]   (fp32 WMMA 16x16x4, wave32)
// block = 128 threads = 4 waves; block tile = 16 rows x 64 cols.
// Fully unrolled K loop: 16x {1 b64 A-load, 2 b32 B-loads, 1 WMMA}, no
// divergence anywhere before the guarded store -> EXEC all-ones at WMMA.
// ---------------------------------------------------------------------------
__global__ void gat_gemm64_wmma(const float* __restrict__ A,
                                const float* __restrict__ W,
                                float* __restrict__ H, int N) {
  const int wave = threadIdx.x >> 5;
  const int lane = threadIdx.x & 31;
  const int half = lane >> 4;     // lane group 0/1: carries K pair {2h,2h+1} of each step
  const int l16  = lane & 15;
  const int m0   = blockIdx.x * 16;

  int row = m0 + l16;
  if (row >= N) row = N - 1;              // clamp once; duplicate compute, store guarded
  const int colB = wave * 16 + l16;

  const float* Arow = A + (size_t)row * 64;

  v8f acc = {};
#pragma unroll
  for (int j = 0; j < 16; ++j) {
    const int ka = 4 * j + 2 * half;      // this lane-group's K pair (even -> 8B aligned)
    const float2 av = *(const float2*)(Arow + ka);
    v2f a; a.x = av.x; a.y = av.y;
    v2f b; b.x = W[ka * 64 + colB]; b.y = W[(ka + 1) * 64 + colB];
    acc = __builtin_amdgcn_wmma_f32_16x16x4_f32(false, a, false, b,
                                                (short)0, acc, false, false);
  }

#pragma unroll
  for (int j = 0; j < 8; ++j) {
    const int r = m0 + j + 8 * half;      // D layout: VGPR j -> M=j (g0) / M=j+8 (g1)
    if (r < N) H[(size_t)r * 64 + colB] = acc[j];
  }
}

// ---------------------------------------------------------------------------
// Layer 0: H[N,64] = x[N,5] @ W0[5,64].  64 MFLOP -> plain FMA kernel;
// x-row loads are wave-uniform (merged), W loads coalesce, 5 FMAs/thread.
// ---------------------------------------------------------------------------
__global__ void gat_gemm_k5(const float* __restrict__ x,
                            const float* __restrict__ W,
                            float* __restrict__ H, int N) {
  const long i = (long)blockIdx.x * blockDim.x + threadIdx.x;   // over N*64
  if (i >= (long)N * 64) return;
  const int n = (int)(i >> 6), c = (int)(i & 63);
  const float* xr = x + (size_t)n * 5;
  float acc = 0.0f;
#pragma unroll
  for (int k = 0; k < 5; ++k) acc += xr[k] * W[k * 64 + c];
  H[i] = acc;
}

// ---------------------------------------------------------------------------
// Per-node pass: alpha_src/alpha_dst dots, init segment max/sum, zero acc.
// One wave per node; lane covers channels {lane, lane+32}.
// ---------------------------------------------------------------------------
__global__ void gat_node_prep(const float* __restrict__ H,
                              const float* __restrict__ a_s,
                              const float* __restrict__ a_d,
                              float* __restrict__ asrc, float* __restrict__ adst,
                              float* __restrict__ m, float* __restrict__ s,
                              float* __restrict__ acc, int N) {
  const int node = blockIdx.x * (blockDim.x >> 5) + (threadIdx.x >> 5);
  if (node >= N) return;
  const int lane = threadIdx.x & 31;
  const size_t base = (size_t)node * 64;

  float h0 = H[base + lane];
  float h1 = H[base + 32 + lane];
  float ps = h0 * a_s[lane] + h1 * a_s[lane + 32];
  float pd = h0 * a_d[lane] + h1 * a_d[lane + 32];
  ps = warp_sum32(ps);
  pd = warp_sum32(pd);
  if (lane == 0) {
    asrc[node] = ps;
    adst[node] = pd;
    m[node] = -__builtin_huge_valf();
    s[node] = 0.0f;
  }
  acc[base + lane] = 0.0f;
  acc[base + 32 + lane] = 0.0f;
}

// ---------------------------------------------------------------------------
// Edge pass A: e = leaky_relu(asrc[src]+adst[dst], 0.2); segment max -> m[dst].
// ---------------------------------------------------------------------------
__global__ void gat_edge_max(const int* __restrict__ ei,
                             const float* __restrict__ asrc,
                             const float* __restrict__ adst,
                             float* __restrict__ ebuf, float* __restrict__ m,
                             int E, int Etot) {
  const int e = blockIdx.x * blockDim.x + threadIdx.x;
  if (e >= Etot) return;
  const int sv = (e < E) ? ei[e]     : (e - E);   // self-loop tail
  const int dv = (e < E) ? ei[E + e] : (e - E);
  float t = asrc[sv] + adst[dv];
  t = (t >= 0.0f) ? t : 0.2f * t;
  ebuf[e] = t;
  atomic_max_f32(&m[dv], t);
}

// ---------------------------------------------------------------------------
// Edge pass B: ex = exp(e - m[dst]); segment sum -> s[dst]; ebuf <- ex.
// ---------------------------------------------------------------------------
__global__ void gat_edge_expsum(const int* __restrict__ ei,
                                const float* __restrict__ m,
                                float* __restrict__ ebuf, float* __restrict__ s,
                                int E, int Etot) {
  const int e = blockIdx.x * blockDim.x + threadIdx.x;
  if (e >= Etot) return;
  const int dv = (e < E) ? ei[E + e] : (e - E);
  float ex = __expf(ebuf[e] - m[dv]);
  ebuf[e] = ex;
  atomicAdd(&s[dv], ex);
}

// ---------------------------------------------------------------------------
// Edge pass C: acc[dst] += h[src] * (ex / s[dst]).  One wave per edge,
// lane covers channels {lane, lane+32}: two coalesced 128B gathers + f32
// atomics; h and acc both fit in the 192MB L2, so atomics resolve there.
// ---------------------------------------------------------------------------
__global__ void gat_edge_scatter(const int* __restrict__ ei,
                                 const float* __restrict__ H,
                                 const float* __restrict__ ebuf,
                                 const float* __restrict__ s,
                                 float* __restrict__ acc, int E, int Etot) {
  const int e = blockIdx.x * (blockDim.x >> 5) + (threadIdx.x >> 5);
  if (e >= Etot) return;
  const int lane = threadIdx.x & 31;
  const int sv = (e < E) ? ei[e]     : (e - E);
  const int dv = (e < E) ? ei[E + e] : (e - E);
  const float alpha = ebuf[e] / s[dv];
  const size_t sb = (size_t)sv * 64, db = (size_t)dv * 64;
  atomicAdd(&acc[db + lane],      H[sb + lane]      * alpha);
  atomicAdd(&acc[db + 32 + lane], H[sb + 32 + lane] * alpha);
}

__global__ void gat_bias_act(float* __restrict__ acc, const float* __restrict__ b,
                             long n64, int relu) {
  const long i = (long)blockIdx.x * blockDim.x + threadIdx.x;
  if (i >= n64) return;
  float v = acc[i] + b[i & 63];
  if (relu) v = fmaxf(v, 0.0f);
  acc[i] = v;
}

__global__ void gat_pool_zero(float* __restrict__ pooled, float* __restrict__ cnt) {
  const int i = blockIdx.x * blockDim.x + threadIdx.x;
  if (i < NUM_GRAPHS * 64) pooled[i] = 0.0f;
  if (i < NUM_GRAPHS) cnt[i] = 0.0f;
}

__global__ void gat_pool_scatter(const float* __restrict__ acc,
                                 const int* __restrict__ batch,
                                 float* __restrict__ pooled, float* __restrict__ cnt,
                                 int N) {
  const int node = blockIdx.x * (blockDim.x >> 5) + (threadIdx.x >> 5);
  if (node >= N) return;
  const int lane = threadIdx.x & 31;
  const int g = batch[node];
  const size_t nb = (size_t)node * 64, gb = (size_t)g * 64;
  atomicAdd(&pooled[gb + lane],      acc[nb + lane]);
  atomicAdd(&pooled[gb + 32 + lane], acc[nb + 32 + lane]);
  if (lane == 0) atomicAdd(&cnt[g], 1.0f);
}

// mean + linear + sigmoid; one wave per graph
__global__ void gat_head(const float* __restrict__ pooled, const float* __restrict__ cnt,
                         const float* __restrict__ lw, const float* __restrict__ lb,
                         float* __restrict__ out) {
  const int g = blockIdx.x * (blockDim.x >> 5) + (threadIdx.x >> 5);
  if (g >= NUM_GRAPHS) return;
  const int lane = threadIdx.x & 31;
  const float inv = 1.0f / fmaxf(cnt[g], 1.0f);
  const size_t gb = (size_t)g * 64;
  float d = pooled[gb + lane] * inv * lw[lane] +
            pooled[gb + 32 + lane] * inv * lw[lane + 32];
  d = warp_sum32(d);
  if (lane == 0) out[g] = 1.0f / (1.0f + __expf(-(d + lb[0])));
}

// ---------------------------------------------------------------------------
extern "C" void kernel_launch(void* const* d_in, const int* in_sizes, int n_in,
                              void* d_out, int out_size, void* d_ws, size_t ws_size,
                              hipStream_t stream) {
  (void)n_in; (void)out_size; (void)ws_size;

  const float* x     = (const float*)d_in[0];
  const int*   ei    = (const int*)d_in[1];
  const int*   batch = (const int*)d_in[2];
  const float* Wm[3] = {(const float*)d_in[3], (const float*)d_in[7],  (const float*)d_in[11]};
  const float* AS[3] = {(const float*)d_in[4], (const float*)d_in[8],  (const float*)d_in[12]};
  const float* AD[3] = {(const float*)d_in[5], (const float*)d_in[9],  (const float*)d_in[13]};
  const float* Bv[3] = {(const float*)d_in[6], (const float*)d_in[10], (const float*)d_in[14]};
  const float* lw = (const float*)d_in[15];
  const float* lb = (const float*)d_in[16];

  const int N    = in_sizes[0] / 5;   // x is [N,5]
  const int E    = in_sizes[1] / 2;   // edge_index is [2,E]
  const int Etot = E + N;             // + self loops

  // carve workspace (256B-aligned slabs)
  uintptr_t p = (uintptr_t)d_ws;
  auto take = [&](size_t bytes) -> float* {
    uintptr_t q = (p + 255) & ~(uintptr_t)255;
    p = q + bytes;
    return (float*)q;
  };
  float* bufA   = take((size_t)N * 64 * sizeof(float));  // h (features)
  float* bufB   = take((size_t)N * 64 * sizeof(float));  // acc / layer output
  float* asrc   = take((size_t)N * sizeof(float));
  float* adst   = take((size_t)N * sizeof(float));
  float* mbuf   = take((size_t)N * sizeof(float));
  float* sbuf   = take((size_t)N * sizeof(float));
  float* ebuf   = take((size_t)Etot * sizeof(float));
  float* pooled = take((size_t)NUM_GRAPHS * 64 * sizeof(float));
  float* cnt    = take((size_t)NUM_GRAPHS * sizeof(float));

  const long n64 = (long)N * 64;

  for (int L = 0; L < 3; ++L) {
    if (L == 0) {
      gat_gemm_k5<<<(int)((n64 + 255) / 256), 256, 0, stream>>>(x, Wm[0], bufA, N);
    } else {
      gat_gemm64_wmma<<<(N + 15) / 16, 128, 0, stream>>>(bufB, Wm[L], bufA, N);
    }
    gat_node_prep<<<(N + 7) / 8, 256, 0, stream>>>(bufA, AS[L], AD[L],
                                                   asrc, adst, mbuf, sbuf, bufB, N);
    gat_edge_max<<<(Etot + 255) / 256, 256, 0, stream>>>(ei, asrc, adst, ebuf, mbuf, E, Etot);
    gat_edge_expsum<<<(Etot + 255) / 256, 256, 0, stream>>>(ei, mbuf, ebuf, sbuf, E, Etot);
    gat_edge_scatter<<<(Etot + 7) / 8, 256, 0, stream>>>(ei, bufA, ebuf, sbuf, bufB, E, Etot);
    gat_bias_act<<<(int)((n64 + 255) / 256), 256, 0, stream>>>(bufB, Bv[L], n64, L < 2);
  }

  gat_pool_zero<<<(NUM_GRAPHS * 64 + 255) / 256, 256, 0, stream>>>(pooled, cnt);
  gat_pool_scatter<<<(N + 7) / 8, 256, 0, stream>>>(bufB, batch, pooled, cnt, N);
  gat_head<<<NUM_GRAPHS / 8, 256, 0, stream>>>(pooled, cnt, lw, lb, (float*)d_out);
}